// WSPContrastiveLoss_36627481101076
// MI455X (gfx1250) — compile-verified
//
#include <hip/hip_runtime.h>
#include <hip/hip_bf16.h>
#include <math.h>

// ---------------------------------------------------------------------------
// WSP contrastive loss, fused for MI455X (gfx1250, wave32, WMMA).
//   sim_Z = (Z Z^T)/T - INF*I,  Z = [norm(z_i); norm(z_j)]  (4096 x 512 fp32)
//   loss  = -(1/N) sum_i ( A_i - lse_i * B_i )
// GEMM via V_WMMA_F32_16X16X4_F32 (full fp32 precision). sim <= 10 for unit
// vectors -> fixed logsumexp shift; all reductions are fixed-order sums.
// ---------------------------------------------------------------------------

#define NROWS   2048
#define DIM     512
#define TWO_N   4096
#define TEMP_INV 10.0f
#define BIG_INF  1.0e8f
#define GAMMA    0.5f        // 1/(2*sigma^2), sigma=1
#define LSE_SHIFT 10.1f

#define ROWS_WG  32          // rows per workgroup (two 16-row WMMA sub-tiles)
#define A_STRIDE 516         // 512 + 4 pad: conflict-free A-fragment reads
#define NWG      (TWO_N / ROWS_WG)   // 128 main workgroups

typedef float v2f __attribute__((ext_vector_type(2)));
typedef float v8f __attribute__((ext_vector_type(8)));

// --------------------------- 1. row L2-normalize ---------------------------
__global__ __launch_bounds__(256)
void k_normalize(const float* __restrict__ zi, const float* __restrict__ zj,
                 float* __restrict__ Zn) {
  const int wave = threadIdx.x >> 5;
  const int lane = threadIdx.x & 31;
  const int row  = blockIdx.x * 8 + wave;           // 512 blocks * 8 waves = 4096
  const float* src = (row < NROWS) ? (zi + (size_t)row * DIM)
                                   : (zj + (size_t)(row - NROWS) * DIM);
  float4 v[4];
  float ss = 0.0f;
#pragma unroll
  for (int q = 0; q < 4; ++q) {
    v[q] = ((const float4*)src)[lane * 4 + q];
    ss += v[q].x * v[q].x + v[q].y * v[q].y + v[q].z * v[q].z + v[q].w * v[q].w;
  }
#pragma unroll
  for (int m = 16; m >= 1; m >>= 1) ss += __shfl_xor(ss, m, 32);
  const float inv = 1.0f / fmaxf(sqrtf(ss), 1e-12f);
  float4* dst = (float4*)(Zn + (size_t)row * DIM);
#pragma unroll
  for (int q = 0; q < 4; ++q) {
    float4 o; o.x = v[q].x * inv; o.y = v[q].y * inv; o.z = v[q].z * inv; o.w = v[q].w * inv;
    dst[lane * 4 + q] = o;
  }
}

// ----------- 2. Sinv[j] = 1 / (rowsum of weights * mask), row j ------------
__global__ __launch_bounds__(256)
void k_weight_rowsum(const float* __restrict__ zpos, const int* __restrict__ labels,
                     float* __restrict__ Sinv) {
  __shared__ float sp[256];
  __shared__ int   sl[256];
  const int j  = blockIdx.x * 256 + threadIdx.x;     // 16 blocks -> 4096 threads
  const float pj = zpos[j & (NROWS - 1)];
  const int   lj = labels[j & (NROWS - 1)];
  float acc = 0.0f;
  for (int k0 = 0; k0 < TWO_N; k0 += 256) {
    const int k = k0 + threadIdx.x;
    sp[threadIdx.x] = zpos[k & (NROWS - 1)];
    sl[threadIdx.x] = labels[k & (NROWS - 1)];
    __syncthreads();
#pragma unroll 8
    for (int t = 0; t < 256; ++t) {
      const float d = pj - sp[t];
      const float w = __expf(-GAMMA * d * d);
      acc += ((lj == sl[t]) && ((k0 + t) != j)) ? w : 0.0f;
    }
    __syncthreads();
  }
  Sinv[j] = 1.0f / acc;
}

// ------------- 3. fused Gram/softmax/weighted-sum per 32-row tile ----------
// Grid: 128 WGs (32 rows each). Block: 256 threads = 8 waves.
// Each wave sweeps column tiles jt = wave, wave+8, ... (16 cols each).
// Every B fragment feeds two WMMAs (row sub-tiles 0-15 and 16-31).
__global__ __launch_bounds__(256)
void k_main(const float* __restrict__ Zn, const float* __restrict__ zpos,
            const int* __restrict__ labels, const float* __restrict__ Sinv,
            float* __restrict__ partial) {
  extern __shared__ float smem[];
  float* buf    = smem;                          // 32*516 A tile; reused for reduction
  float* red2   = smem + ROWS_WG * A_STRIDE;     // 32*8*3
  float* rowval = red2 + ROWS_WG * 8 * 3;        // 32

  const int tid     = threadIdx.x;
  const int wave    = tid >> 5;
  const int lane    = tid & 31;
  const int lane_lo = lane & 15;
  const int hi      = lane >> 4;                 // K half-select / row+8 select
  const int i0      = blockIdx.x * ROWS_WG;

  // Stage A tile (rows i0..i0+31) into LDS, padded stride.
  for (int idx = tid; idx < ROWS_WG * (DIM / 4); idx += 256) {
    const int r  = idx >> 7;                     // / (DIM/4)
    const int c4 = idx & 127;
    float4 v = ((const float4*)(Zn + (size_t)(i0 + r) * DIM))[c4];
    float* d = &buf[r * A_STRIDE + c4 * 4];
    d[0] = v.x; d[1] = v.y; d[2] = v.z; d[3] = v.w;
  }
  __syncthreads();

  // Metadata for the 16 rows this thread accumulates: m = blk*16 + r + 8*hi.
  float p_row[16]; int l_row[16];
#pragma unroll
  for (int m = 0; m < 16; ++m) {
    const int i = i0 + ((m & 8) << 1) + (m & 7) + 8 * hi;  // blk=(m>>3), r=(m&7)
    p_row[m] = zpos[i & (NROWS - 1)];
    l_row[m] = labels[i & (NROWS - 1)];
  }

  float sumE[16], sumA[16], sumB[16];
#pragma unroll
  for (int m = 0; m < 16; ++m) { sumE[m] = 0.f; sumA[m] = 0.f; sumB[m] = 0.f; }

  const float* ap0 = &buf[lane_lo * A_STRIDE + 2 * hi];                // rows 0-15
  const float* ap1 = &buf[(16 + lane_lo) * A_STRIDE + 2 * hi];         // rows 16-31

  for (int jt = wave; jt < TWO_N / 16; jt += 8) {
    const int j0 = jt * 16;
    const int j  = j0 + lane_lo;                           // this lane's C column
    const float* bp = Zn + (size_t)j * DIM + 2 * hi;       // B fragment source

    if (j0 + 128 < TWO_N)                                  // prefetch next B tile
      __builtin_prefetch(Zn + (size_t)(j0 + 128 + lane_lo) * DIM, 0, 0);

    v8f acc0 = {}, acc1 = {};
    for (int k = 0; k < DIM; k += 16) {                    // 4 K-steps per chunk
      v2f b[4], a0[4], a1[4];
#pragma unroll
      for (int t = 0; t < 4; ++t) b[t]  = *(const v2f*)(bp  + k + 4 * t);
#pragma unroll
      for (int t = 0; t < 4; ++t) a0[t] = *(const v2f*)(ap0 + k + 4 * t);
#pragma unroll
      for (int t = 0; t < 4; ++t) a1[t] = *(const v2f*)(ap1 + k + 4 * t);
#pragma unroll
      for (int t = 0; t < 4; ++t) {
        acc0 = __builtin_amdgcn_wmma_f32_16x16x4_f32(false, a0[t], false, b[t],
                                                     (short)0, acc0, false, false);
        acc1 = __builtin_amdgcn_wmma_f32_16x16x4_f32(false, a1[t], false, b[t],
                                                     (short)0, acc1, false, false);
      }
    }

    const float pj   = zpos[j & (NROWS - 1)];
    const int   lj   = labels[j & (NROWS - 1)];
    const float iSj  = Sinv[j];

#pragma unroll
    for (int blk = 0; blk < 2; ++blk) {
#pragma unroll
      for (int r = 0; r < 8; ++r) {
        const int m = blk * 8 + r;                         // metadata slot
        const int i = i0 + blk * 16 + r + 8 * hi;          // global row
        const float dot = blk ? acc1[r] : acc0[r];
        const float sim = dot * TEMP_INV - ((i == j) ? BIG_INF : 0.0f);
        sumE[m] += __expf(sim - LSE_SHIFT);                // diag underflows to 0
        const float d = p_row[m] - pj;
        const float w = __expf(-GAMMA * d * d) * iSj;      // unconditional: no exec branch
        const float fw = ((l_row[m] == lj) && (i != j)) ? w : 0.0f;
        sumA[m] += sim * fw;
        sumB[m] += fw;
      }
    }
  }

  __syncthreads();                                         // A tile dead; reuse buf

  // Per-thread partials -> buf[q*4096 + row*128 + slot], row = blk*16+r+8*hi
  const int slot = wave * 16 + lane_lo;
#pragma unroll
  for (int m = 0; m < 16; ++m) {
    const int row = ((m & 8) << 1) + (m & 7) + 8 * hi;
    buf[0 * 4096 + row * 128 + slot] = sumE[m];
    buf[1 * 4096 + row * 128 + slot] = sumA[m];
    buf[2 * 4096 + row * 128 + slot] = sumB[m];
  }
  __syncthreads();

  { // stage 2: 8 threads per row (32 rows), fixed-order sums
    const int row = tid >> 3, part = tid & 7;
    float e = 0.f, a = 0.f, b = 0.f;
    for (int s = part; s < 128; s += 8) {
      e += buf[0 * 4096 + row * 128 + s];
      a += buf[1 * 4096 + row * 128 + s];
      b += buf[2 * 4096 + row * 128 + s];
    }
    red2[(row * 8 + part) * 3 + 0] = e;
    red2[(row * 8 + part) * 3 + 1] = a;
    red2[(row * 8 + part) * 3 + 2] = b;
  }
  __syncthreads();

  if (tid < ROWS_WG) {
    float e = 0.f, a = 0.f, b = 0.f;
    for (int part = 0; part < 8; ++part) {
      e += red2[(tid * 8 + part) * 3 + 0];
      a += red2[(tid * 8 + part) * 3 + 1];
      b += red2[(tid * 8 + part) * 3 + 2];
    }
    const float lse = LSE_SHIFT + logf(e);
    rowval[tid] = a - lse * b;
  }
  __syncthreads();
  if (tid == 0) {
    float t = 0.f;
    for (int m = 0; m < ROWS_WG; ++m) t += rowval[m];
    partial[blockIdx.x] = t;
  }
}

// ------------------------- 4. final scalar reduce --------------------------
__global__ __launch_bounds__(256)
void k_final(const float* __restrict__ partial, float* __restrict__ out) {
  __shared__ float s[256];
  s[threadIdx.x] = (threadIdx.x < NWG) ? partial[threadIdx.x] : 0.0f;
  __syncthreads();
  for (int off = 128; off >= 1; off >>= 1) {
    if (threadIdx.x < off) s[threadIdx.x] += s[threadIdx.x + off];
    __syncthreads();
  }
  if (threadIdx.x == 0) out[0] = -s[0] / (float)NROWS;
}

// ---------------------------------------------------------------------------
extern "C" void kernel_launch(void* const* d_in, const int* in_sizes, int n_in,
                              void* d_out, int out_size, void* d_ws, size_t ws_size,
                              hipStream_t stream) {
  const float* z_i    = (const float*)d_in[0];   // [2048,512] f32
  const float* z_j    = (const float*)d_in[1];   // [2048,512] f32
  const int*   labels = (const int*)d_in[2];     // [2048] i32
  const float* z_pos  = (const float*)d_in[3];   // [2048,1] f32
  float* out = (float*)d_out;

  // workspace layout: Zn (8 MB) | Sinv (16 KB) | partial (512 B)
  float* Zn      = (float*)d_ws;
  float* Sinv    = Zn + (size_t)TWO_N * DIM;
  float* partial = Sinv + TWO_N;

  // dynamic LDS for k_main: A tile + stage-2 scratch
  const size_t lds_bytes = (ROWS_WG * A_STRIDE + ROWS_WG * 8 * 3 + ROWS_WG) * sizeof(float);

  k_normalize    <<<TWO_N / 8, 256, 0, stream>>>(z_i, z_j, Zn);
  k_weight_rowsum<<<TWO_N / 256, 256, 0, stream>>>(z_pos, labels, Sinv);
  k_main         <<<NWG, 256, lds_bytes, stream>>>(Zn, z_pos, labels, Sinv, partial);
  k_final        <<<1, 256, 0, stream>>>(partial, out);
}